// Attention_12558484373680
// MI455X (gfx1250) — compile-verified
//
#include <hip/hip_runtime.h>
#include <hip/hip_bf16.h>
#include <math.h>

// Problem constants (from the reference)
#define SQ 4096
#define BQ 128
#define HQ 128

typedef __attribute__((ext_vector_type(2))) float v2f;
typedef __attribute__((ext_vector_type(8))) float v8f;

// ---------------------------------------------------------------------------
// float atomic max via monotone integer mapping:
//  - positive floats compare correctly as signed int (use atomicMax on int)
//  - negative floats compare reversed as unsigned (use atomicMin on uint)
// init value must be -inf (0xFF800000).
// ---------------------------------------------------------------------------
__device__ __forceinline__ void atomicMaxF(float* addr, float val) {
    if (val >= 0.0f) {
        atomicMax((int*)addr, __float_as_int(val));
    } else {
        atomicMin((unsigned int*)addr, __float_as_uint(val));
    }
}

// ---------------------------------------------------------------------------
// Init per-column softmax stats (must run every call: graph replays don't
// re-poison, and we must be deterministic).
// ---------------------------------------------------------------------------
__global__ void init_stats_kernel(float* __restrict__ m, float* __restrict__ Z) {
    int b = threadIdx.x;
    if (b < BQ) {
        m[b] = -INFINITY;
        Z[b] = 0.0f;
    }
}

// ---------------------------------------------------------------------------
// Generic fp32 WMMA GEMM: C[M,N] = op(A[M,K] @ B[K,N]), row-major, one wave
// per 16x16 tile, K stepped by 4 with V_WMMA_F32_16X16X4_F32.
// Fragment layouts per CDNA5 ISA 7.12.2:
//   A 16x4 : lane l (0..15)=row M; VGPR j + 16-lane half selects K = 2*half+j
//   B 4x16 : lane l = col N;        VGPR j + half selects K = 2*half+j
//   C 16x16: VGPR r, half 0 -> M=r, half 1 -> M=r+8; lane l = N
// Used for q = d@W_a, t = tanh(d@W_p), out = tanh(x@W_c) -- all tiny
// (<0.1% of runtime) but exercise the matrix pipe in full fp32.
// ---------------------------------------------------------------------------
__global__ void wmma_gemm_f32(const float* __restrict__ A,
                              const float* __restrict__ Bm,
                              float* __restrict__ C,
                              int K, int N, int applyTanh) {
    const int lane = threadIdx.x & 31;
    const int l    = lane & 15;
    const int half = lane >> 4;
    const int mi   = blockIdx.y * 16;
    const int ni   = blockIdx.x * 16;

    v8f c = {};
    for (int k0 = 0; k0 < K; k0 += 4) {
        const int ka = k0 + 2 * half;
        v2f a, b;
        a[0] = A[(size_t)(mi + l) * K + ka];
        a[1] = A[(size_t)(mi + l) * K + ka + 1];
        b[0] = Bm[(size_t)ka * N + ni + l];
        b[1] = Bm[(size_t)(ka + 1) * N + ni + l];
        c = __builtin_amdgcn_wmma_f32_16x16x4_f32(false, a, false, b,
                                                  (short)0, c, false, false);
    }
#pragma unroll
    for (int r = 0; r < 8; ++r) {
        float v = c[r];
        if (applyTanh) v = tanhf(v);
        C[(size_t)(mi + r + 8 * half) * N + ni + l] = v;
    }
}

// ---------------------------------------------------------------------------
// p_t[b] = S * sigmoid( dot(t[b,:], v_p) ), t = tanh(d@W_p) precomputed.
// One block per b, LDS tree reduce over H=128.
// ---------------------------------------------------------------------------
__global__ void pt_kernel(const float* __restrict__ t,
                          const float* __restrict__ v_p,
                          float* __restrict__ pt) {
    __shared__ float sd[HQ];
    const int b = blockIdx.x;
    const int h = threadIdx.x;
    sd[h] = t[(size_t)b * HQ + h] * v_p[h];
    __syncthreads();
    for (int o = HQ / 2; o > 0; o >>= 1) {
        if (h < o) sd[h] += sd[h + o];
        __syncthreads();
    }
    if (h == 0) {
        float x = sd[0];
        pt[b] = (float)SQ * (1.0f / (1.0f + expf(-x)));
    }
}

// ---------------------------------------------------------------------------
// THE bandwidth kernel: streams all 256 MB of e exactly once.
// scoresT[b,s] = e[s,b,:] . q[b,:]   (transposed scratch for L2-friendly
// softmax passes), plus per-column running max via one atomic per wave.
//
// Layout: block = 256 threads = 8 waves; wave w owns column b = by*8+w and
// 128 consecutive s. Each 16-lane half-wave processes one s per step: lane
// reads 32 contiguous bytes (2x float4) => each half-wave issues one fully
// coalesced 512 B row read. 8 FMA + 4-step butterfly per row keeps VALU work
// ~10 instructions / 512 B, safely under the issue budget for 23.3 TB/s.
// Results are banked into one lane per s so the scoresT store is a single
// 128 B-line write per 32 s.
// ---------------------------------------------------------------------------
__global__ void scores_kernel(const float* __restrict__ e,
                              const float* __restrict__ q,
                              float* __restrict__ scoresT,
                              float* __restrict__ m) {
    const int lane = threadIdx.x & 31;
    const int wave = threadIdx.x >> 5;
    const int b    = blockIdx.y * 8 + wave;
    const int s0   = blockIdx.x * 128;
    const int l16  = lane & 15;
    const int half = lane >> 4;

    const float4 qa = *(const float4*)(q + (size_t)b * HQ + l16 * 8);
    const float4 qb = *(const float4*)(q + (size_t)b * HQ + l16 * 8 + 4);

    float maxv = -INFINITY;

    for (int gi = 0; gi < 4; ++gi) {
        float mine = 0.0f;
#pragma unroll
        for (int i = 0; i < 16; ++i) {
            const int s = s0 + gi * 32 + 2 * i + half;
            const float* row = e + ((size_t)s * BQ + b) * HQ + l16 * 8;
            const float4 ea = *(const float4*)(row);
            const float4 eb = *(const float4*)(row + 4);
            float sum = ea.x * qa.x + ea.y * qa.y + ea.z * qa.z + ea.w * qa.w
                      + eb.x * qb.x + eb.y * qb.y + eb.z * qb.z + eb.w * qb.w;
            // reduce across the 16-lane half (xor 1,2,4,8 stays in-half)
            sum += __shfl_xor(sum, 1);
            sum += __shfl_xor(sum, 2);
            sum += __shfl_xor(sum, 4);
            sum += __shfl_xor(sum, 8);
            maxv = fmaxf(maxv, sum);
            if (l16 == i) mine = sum;   // bank for coalesced store
        }
        scoresT[(size_t)b * SQ + s0 + gi * 32 + 2 * l16 + half] = mine;
    }

    maxv = fmaxf(maxv, __shfl_xor(maxv, 16));  // merge the two halves
    if (lane == 0) atomicMaxF(&m[b], maxv);
}

// ---------------------------------------------------------------------------
// exp pass: scoresT[b,s] <- exp(scoresT[b,s] - m[b]); Z[b] += block sum.
// 2 MB scratch -> entirely L2 resident, coalesced along s.
// grid = (S/1024, B), block = 256, 4 elements/thread.
// ---------------------------------------------------------------------------
__global__ void expsum_kernel(float* __restrict__ scoresT,
                              const float* __restrict__ m,
                              float* __restrict__ Z) {
    __shared__ float sd[256];
    const int tid = threadIdx.x;
    const int b   = blockIdx.y;
    const float mb = m[b];
    float local = 0.0f;
#pragma unroll
    for (int j = 0; j < 4; ++j) {
        const size_t idx = (size_t)b * SQ + blockIdx.x * 1024 + j * 256 + tid;
        const float v = expf(scoresT[idx] - mb);
        scoresT[idx] = v;
        local += v;
    }
    sd[tid] = local;
    __syncthreads();
    for (int o = 128; o > 0; o >>= 1) {
        if (tid < o) sd[tid] += sd[tid + o];
        __syncthreads();
    }
    if (tid == 0) atomicAdd(&Z[b], sd[0]);
}

// ---------------------------------------------------------------------------
// w[s,b] = (exp/Z) * gauss, gauss = exp(-diff^2/2) hard-masked |diff|<=2
// (SIGMA = D_WIN/2 = 1 -> 2*sigma^2 = 2). Writes coalesced into d_out[B*H..].
// Reads of scoresT are strided but hit L2.
// ---------------------------------------------------------------------------
__global__ void w_kernel(const float* __restrict__ scoresT,
                         const float* __restrict__ Z,
                         const float* __restrict__ pt,
                         float* __restrict__ w_out) {
    const int i = blockIdx.x * 256 + threadIdx.x;   // i = s*B + b
    const int s = i >> 7;       // /BQ
    const int b = i & (BQ - 1);
    const float ex   = scoresT[(size_t)b * SQ + s];
    const float diff = pt[b] - (float)s;
    const float g    = (fabsf(diff) <= 2.0f) ? expf(-0.5f * diff * diff) : 0.0f;
    w_out[i] = ex * g / Z[b];
}

// ---------------------------------------------------------------------------
// Windowed context: the Gaussian mask zeroes everything outside
// s in [p_t-2, p_t+2] (<=5 rows), so context is a tiny gather instead of a
// second 256 MB pass over e. Also builds x = [context, d] for the final GEMM.
// One block per b, thread = h.
// ---------------------------------------------------------------------------
__global__ void context_kernel(const float* __restrict__ e,
                               const float* __restrict__ d,
                               const float* __restrict__ w,
                               const float* __restrict__ pt,
                               float* __restrict__ x) {
    const int b = blockIdx.x;
    const int h = threadIdx.x;
    const float p = pt[b];
    int lo = (int)ceilf(p - 2.0f); if (lo < 0) lo = 0;
    int hi = (int)floorf(p + 2.0f); if (hi > SQ - 1) hi = SQ - 1;
    float acc = 0.0f;
    for (int s = lo; s <= hi; ++s) {
        acc += e[((size_t)s * BQ + b) * HQ + h] * w[(size_t)s * BQ + b];
    }
    x[(size_t)b * (2 * HQ) + h]      = acc;            // context
    x[(size_t)b * (2 * HQ) + HQ + h] = d[(size_t)b * HQ + h];  // concat d
}

// ---------------------------------------------------------------------------
extern "C" void kernel_launch(void* const* d_in, const int* in_sizes, int n_in,
                              void* d_out, int out_size, void* d_ws, size_t ws_size,
                              hipStream_t stream) {
    const float* e   = (const float*)d_in[0];   // [S,B,H]
    const float* d   = (const float*)d_in[1];   // [1,B,H] -> [B,H]
    const float* W_a = (const float*)d_in[2];   // [H,H]
    const float* W_p = (const float*)d_in[3];   // [H,H]
    const float* v_p = (const float*)d_in[4];   // [H,1]
    const float* W_c = (const float*)d_in[5];   // [2H,H]
    float* out = (float*)d_out;                 // [B*H] output ++ [S*B] w

    // workspace carve-up (floats): ~2.4 MB total
    float* ws      = (float*)d_ws;
    float* q       = ws;                        // B*H   = 16384
    float* t       = ws + 16384;                // B*H   = 16384
    float* pt      = ws + 32768;                // B     = 128
    float* m       = ws + 32896;                // B     = 128
    float* Z       = ws + 33024;                // B     = 128
    float* scoresT = ws + 33152;                // B*S   = 524288
    float* x       = ws + 33152 + (size_t)BQ * SQ;  // B*2H = 32768

    float* w_out = out + BQ * HQ;               // w region of d_out

    // 1. softmax stats init (must happen every call)
    init_stats_kernel<<<1, 128, 0, stream>>>(m, Z);

    // 2. q = d @ W_a   (WMMA fp32)
    wmma_gemm_f32<<<dim3(HQ / 16, BQ / 16), 32, 0, stream>>>(d, W_a, q, HQ, HQ, 0);

    // 3. t = tanh(d @ W_p)   (WMMA fp32, fused tanh)
    wmma_gemm_f32<<<dim3(HQ / 16, BQ / 16), 32, 0, stream>>>(d, W_p, t, HQ, HQ, 1);

    // 4. p_t[b]
    pt_kernel<<<BQ, HQ, 0, stream>>>(t, v_p, pt);

    // 5. scores + column max: the single 256 MB streaming pass over e
    scores_kernel<<<dim3(SQ / 128, BQ / 8), 256, 0, stream>>>(e, q, scoresT, m);

    // 6. exp + Z (L2-resident)
    expsum_kernel<<<dim3(SQ / 1024, BQ), 256, 0, stream>>>(scoresT, m, Z);

    // 7. w = softmax * gauss -> d_out
    w_kernel<<<(SQ * BQ) / 256, 256, 0, stream>>>(scoresT, Z, pt, w_out);

    // 8. windowed context + concat -> x
    context_kernel<<<BQ, HQ, 0, stream>>>(e, d, w_out, pt, x);

    // 9. output = tanh(x @ W_c) -> d_out  (WMMA fp32, K=256)
    wmma_gemm_f32<<<dim3(HQ / 16, BQ / 16), 32, 0, stream>>>(x, W_c, out, 2 * HQ, HQ, 1);
}